// OrthogonalLayer1D_63823214018735
// MI455X (gfx1250) — compile-verified
//
#include <hip/hip_runtime.h>
#include <hip/hip_bf16.h>

#ifndef __has_builtin
#define __has_builtin(x) 0
#endif

#if __has_builtin(__builtin_amdgcn_global_load_async_to_lds_b128) && \
    __has_builtin(__builtin_amdgcn_s_wait_asynccnt)
#define HAS_ASYNC_LDS 1
#else
#define HAS_ASYNC_LDS 0
#endif

#if __has_builtin(__builtin_amdgcn_wmma_f32_16x16x4_f32)
#define HAS_WMMA 1
#else
#define HAS_WMMA 0
#endif

typedef __attribute__((ext_vector_type(2))) float v2f;
typedef __attribute__((ext_vector_type(8))) float v8f;
typedef __attribute__((ext_vector_type(4))) int   v4i;

typedef __attribute__((address_space(1))) v4i gbl_v4i;  // global
typedef __attribute__((address_space(3))) v4i lds_v4i;  // LDS

#define EPS   1e-8f
#define NM    8      // n_models
#define BATCH 4096
#define DIM   1024
#define TPB   256    // 8 waves; DIM/TPB = 4 floats per thread

// ---- async global -> LDS copy of 16B per lane (CDNA5 ASYNCcnt path) -------
__device__ __forceinline__ void async_cp16(float4* lds_dst, const float* gsrc) {
#if HAS_ASYNC_LDS
  __builtin_amdgcn_global_load_async_to_lds_b128(
      (gbl_v4i*)(v4i*)(void*)gsrc,
      (lds_v4i*)(v4i*)(void*)lds_dst,
      /*imm offset*/ 0, /*cpol*/ 0);
#else
  *lds_dst = *reinterpret_cast<const float4*>(gsrc);
#endif
}

// ---- 32-lane sum; WMMA f32 16x16x4 with B = ones does the heavy lifting ---
__device__ __forceinline__ float wave_sum(float p) {
#if HAS_WMMA
  v2f a;  a[0] = p;    a[1] = 0.0f;   // A[m,k]: lanes carry K=0/K=2; K=1,3 = 0
  v2f bo; bo[0] = 1.0f; bo[1] = 1.0f; // B = all ones -> C[m,n] = sum_k A[m,k]
  v8f c = {};
  c = __builtin_amdgcn_wmma_f32_16x16x4_f32(false, a, false, bo,
                                            (short)0, c, false, false);
  float s = 0.0f;
#pragma unroll
  for (int k = 0; k < 8; ++k) s += c[k];   // lanes<16: rows 0-7, lanes>=16: rows 8-15
  s += __shfl_xor(s, 16, 32);              // combine the two halves
  return s;
#else
#pragma unroll
  for (int off = 16; off > 0; off >>= 1) p += __shfl_xor(p, off, 32);
  return p;
#endif
}

// ---- 256-thread block sum (one barrier; caller double-buffers red8) -------
__device__ __forceinline__ float block_sum(float p, float* red8, int tid) {
  float w = wave_sum(p);
  if ((tid & 31) == 0) red8[tid >> 5] = w;
  __syncthreads();
  float s = 0.0f;
#pragma unroll
  for (int k = 0; k < TPB / 32; ++k) s += red8[k];
  return s;
}

__global__ __launch_bounds__(TPB) void mgs8_kernel(const float* __restrict__ x,
                                                   float* __restrict__ out) {
  const int b   = blockIdx.x;   // batch element
  const int tid = threadIdx.x;  // owns elements [4*tid, 4*tid+4)

  __shared__ float4 stage[2][TPB];      // double-buffered 4KB model slice
  __shared__ float  red[2][TPB / 32];   // double-buffered reduction scratch

  const size_t slice = (size_t)BATCH * DIM;            // stride between models
  const size_t base  = (size_t)b * DIM + 4 * (size_t)tid;

  float4 q[NM];   // basis chunks live in registers (32 VGPRs)
  int rb = 0;

  // prefetch model 0
  async_cp16(&stage[0][tid], x + base);

#pragma unroll
  for (int i = 0; i < NM; ++i) {
    // prefetch next model while this one is being orthogonalized
    if (i + 1 < NM) {
      async_cp16(&stage[(i + 1) & 1][tid], x + base + (size_t)(i + 1) * slice);
#if HAS_ASYNC_LDS
      __builtin_amdgcn_s_wait_asynccnt(1);   // slice i is complete (in-order)
#endif
    } else {
#if HAS_ASYNC_LDS
      __builtin_amdgcn_s_wait_asynccnt(0);
#endif
    }
#if HAS_ASYNC_LDS
    asm volatile("" ::: "memory");  // keep the ds read below the async wait
#endif
    // each thread reads only its own lane's 16B -> same-wave wait suffices
    float4 v = stage[i & 1][tid];

    if (i > 0) {
      // double-pass MGS against q[0..i-1] (q[j] already unit; safe_unit is idempotent)
#pragma unroll
      for (int pass = 0; pass < 2; ++pass) {
#pragma unroll
        for (int j = 0; j < i; ++j) {
          float pd = v.x * q[j].x + v.y * q[j].y + v.z * q[j].z + v.w * q[j].w;
          float d  = block_sum(pd, red[rb], tid); rb ^= 1;
          v.x -= d * q[j].x; v.y -= d * q[j].y;
          v.z -= d * q[j].z; v.w -= d * q[j].w;
        }
      }
    }

    // normalize
    float pn = v.x * v.x + v.y * v.y + v.z * v.z + v.w * v.w;
    float nn = block_sum(pn, red[rb], tid); rb ^= 1;
    float n  = sqrtf(nn);
    if (n > EPS) {                 // max(n, EPS) == n on this path
      float inv = 1.0f / n;
      v.x *= inv; v.y *= inv; v.z *= inv; v.w *= inv;
    } else if (i > 0) {            // degenerate fallback: e0
      v.x = (tid == 0) ? 1.0f : 0.0f; v.y = 0.0f; v.z = 0.0f; v.w = 0.0f;
    }                              // i == 0: safe_unit leaves v unchanged

    q[i] = v;
    *reinterpret_cast<float4*>(out + base + (size_t)i * slice) = v;
  }
}

extern "C" void kernel_launch(void* const* d_in, const int* in_sizes, int n_in,
                              void* d_out, int out_size, void* d_ws, size_t ws_size,
                              hipStream_t stream) {
  (void)in_sizes; (void)n_in; (void)d_ws; (void)ws_size; (void)out_size;
  const float* x   = (const float*)d_in[0];  // [8, 4096, 1024] f32
  float*       out = (float*)d_out;          // [8, 4096, 1024] f32
  mgs8_kernel<<<dim3(BATCH), dim3(TPB), 0, stream>>>(x, out);
}